// LabelSmoothingCrossEntropy_Weight1_10934986736275
// MI455X (gfx1250) — compile-verified
//
#include <hip/hip_runtime.h>
#include <stdint.h>

// ---------------------------------------------------------------------------
// LabelSmoothingCrossEntropy + weight-max loss, MI455X (gfx1250, wave32).
//
// Memory-bound: weight[256,512,512] f32 = 256MB streamed once (-> TDM async
// tiles into LDS, double buffered), logits[256,32000] f32 = 32MB (L2-resident
// across replays, plain b128 loads). HBM roofline ~12.4us @ 23.3 TB/s.
// No matmul in this workload -> WMMA inapplicable; the CDNA5-specific path is
// the Tensor Data Mover (tensor_load_to_lds + s_wait_tensorcnt).
// ---------------------------------------------------------------------------

#define BV 32000
#define BB 256
#define NN 512
#define WTROWS 32                      // tile rows per TDM transfer
#define WTILE_ELEMS (WTROWS * NN)      // 16384 floats = 64 KB
#define NTILES (NN / WTROWS)           // 16 tiles per batch slice

typedef unsigned int u32x4 __attribute__((ext_vector_type(4)));
typedef int          i32x4 __attribute__((ext_vector_type(4)));
typedef int          i32x8 __attribute__((ext_vector_type(8)));
typedef float        f32x4 __attribute__((ext_vector_type(4)));

#if defined(__gfx1250__) && __has_builtin(__builtin_amdgcn_tensor_load_to_lds)
#define USE_TDM 1
#else
#define USE_TDM 0
#endif

#if USE_TDM
// Issue one TDM load: 2D tile (tile_dim0=512 elems contiguous, tile_dim1=32
// rows, row stride 512 elems, data_size=4B) from global -> LDS byte offset.
// D# bit packing per CDNA5 ISA section 8.3 (group0) / 8.4 (group1).
__device__ __forceinline__ void tdm_load_tile(const float* gptr, uint32_t lds_byte_off) {
  uint64_t ga = (uint64_t)(uintptr_t)gptr;
  u32x4 g0;
  g0[0] = 1u;                                          // count=1 (valid), user mode
  g0[1] = lds_byte_off;                                // lds_addr [63:32]
  g0[2] = (uint32_t)ga;                                // global_addr[31:0]
  g0[3] = (uint32_t)((ga >> 32) & 0x1FFFFFFu)          // global_addr[56:32]
          | 0x80000000u;                               // type=2 ("image") [127:126]
  i32x8 g1;
  g1[0] = (int)(2u << 16);          // data_size=2 (4 bytes); wg_mask=0 (not in cluster)
  g1[1] = (int)(512u << 16);        // tensor_dim0 = 512 (low 16 bits at [63:48])
  g1[2] = (int)(512u << 16);        // tensor_dim1 = 512 (low 16 bits at dw2[31:16])
  g1[3] = (int)(512u << 16);        // tile_dim0 = 512   [127:112]
  g1[4] = (int)WTROWS;              // tile_dim1 = 32    [143:128]; tile_dim2=0
  g1[5] = 512;                      // tensor_dim0_stride = 512 elems [191:160]
  g1[6] = 0;
  g1[7] = 0;
  i32x4 gz = {0, 0, 0, 0};          // groups 2/3 unused (2D tensor)
#if __has_include(<hip/amd_detail/amd_gfx1250_TDM.h>)
  i32x8 gz8 = {0, 0, 0, 0, 0, 0, 0, 0};
  __builtin_amdgcn_tensor_load_to_lds(g0, g1, gz, gz, gz8, 0);   // clang-23 arity
#else
  __builtin_amdgcn_tensor_load_to_lds(g0, g1, gz, gz, 0);        // ROCm 7.2 arity
#endif
}
#endif

// ---------------------------------------------------------------------------
// Kernel 1: per-row label-smoothed CE loss (one block / row).
// loss_b = lse - 0.9*x[target] - 0.1*mean(x)
// ---------------------------------------------------------------------------
__global__ void ce_kernel(const float* __restrict__ logits,
                          const int* __restrict__ target,
                          float* __restrict__ row_loss) {
  const int b = blockIdx.x;
  const float* row = logits + (size_t)b * BV;
  const f32x4* row4 = (const f32x4*)row;
  const int nvec = BV / 4;                 // 8000

  float m = -INFINITY, s = 0.0f, t = 0.0f;
  for (int j = threadIdx.x; j < nvec; j += blockDim.x) {
    f32x4 v = row4[j];
    float mx = fmaxf(fmaxf(v[0], v[1]), fmaxf(v[2], v[3]));
    float M  = fmaxf(m, mx);
    s = s * __expf(m - M)
        + __expf(v[0] - M) + __expf(v[1] - M)
        + __expf(v[2] - M) + __expf(v[3] - M);
    m = M;
    t += (v[0] + v[1]) + (v[2] + v[3]);
  }
  // wave32 reduce (online-softmax merge)
  #pragma unroll
  for (int off = 16; off > 0; off >>= 1) {
    float om = __shfl_xor(m, off, 32);
    float os = __shfl_xor(s, off, 32);
    float ot = __shfl_xor(t, off, 32);
    float M = fmaxf(m, om);
    s = s * __expf(m - M) + os * __expf(om - M);
    m = M;
    t += ot;
  }
  __shared__ float sm[8], ss[8], st[8];
  const int wave = threadIdx.x >> 5, lane = threadIdx.x & 31;
  if (lane == 0) { sm[wave] = m; ss[wave] = s; st[wave] = t; }
  __syncthreads();
  if (threadIdx.x == 0) {
    float M = sm[0], S = ss[0], T = st[0];
    #pragma unroll
    for (int w = 1; w < 8; ++w) {
      float M2 = fmaxf(M, sm[w]);
      S = S * __expf(M - M2) + ss[w] * __expf(sm[w] - M2);
      M = M2;
      T += st[w];
    }
    const float lse = M + __logf(S);
    const float xt  = row[target[b]];
    row_loss[b] = lse - 0.9f * xt - 0.1f * (T * (1.0f / (float)BV));
  }
}

// ---------------------------------------------------------------------------
// Kernel 2: per-batch max over weight[b] with diagonal sign flipped.
// TDM double-buffered global->LDS streaming; 512 threads (16 waves) reduce.
// ---------------------------------------------------------------------------
__global__ void wmax_kernel(const float* __restrict__ weight,
                            float* __restrict__ wmax_out) {
  extern __shared__ __align__(16) float smem[];      // 2 * 64KB tiles
  const int b   = blockIdx.x;
  const int tid = threadIdx.x;                       // 0..511
  const int wave = tid >> 5;
  const float* slice = weight + (size_t)b * NN * NN;

  float lmax = -INFINITY;

#if USE_TDM
  const uint32_t lds_base = (uint32_t)(uintptr_t)smem;   // LDS byte offset
  if (wave == 0) {
    tdm_load_tile(slice, lds_base);                      // prefetch tile 0
  }
  for (int tI = 0; tI < NTILES; ++tI) {
    if (wave == 0) {
      if (tI + 1 < NTILES) {
        tdm_load_tile(slice + (size_t)(tI + 1) * WTILE_ELEMS,
                      lds_base + (uint32_t)(((tI + 1) & 1) * WTILE_ELEMS * 4));
        // in-order TDM: <=1 outstanding means tile tI has landed
        __builtin_amdgcn_s_wait_tensorcnt(1);
      } else {
        __builtin_amdgcn_s_wait_tensorcnt(0);
      }
    }
    __syncthreads();                                     // tile tI visible to all waves
    const f32x4* buf4 = (const f32x4*)(smem + (tI & 1) * WTILE_ELEMS);
    const int r0 = tI * WTROWS;
    #pragma unroll
    for (int j = tid; j < WTILE_ELEMS / 4; j += 512) {
      f32x4 v = buf4[j];
      const int row = r0 + (j >> 7);                     // 128 float4 per row
      const int c0  = (j & 127) << 2;
      if (c0 + 0 == row) v[0] = -v[0];
      if (c0 + 1 == row) v[1] = -v[1];
      if (c0 + 2 == row) v[2] = -v[2];
      if (c0 + 3 == row) v[3] = -v[3];
      lmax = fmaxf(lmax, fmaxf(fmaxf(v[0], v[1]), fmaxf(v[2], v[3])));
    }
    __syncthreads();                                     // buffer free for tile tI+2
  }
#else
  // Fallback: non-temporal b128 streaming straight from HBM.
  const f32x4* s4 = (const f32x4*)slice;
  for (int j = tid; j < NN * NN / 4; j += 512) {
    f32x4 v = __builtin_nontemporal_load(&s4[j]);
    const int row = j >> 7;
    const int c0  = (j & 127) << 2;
    if (c0 + 0 == row) v[0] = -v[0];
    if (c0 + 1 == row) v[1] = -v[1];
    if (c0 + 2 == row) v[2] = -v[2];
    if (c0 + 3 == row) v[3] = -v[3];
    lmax = fmaxf(lmax, fmaxf(fmaxf(v[0], v[1]), fmaxf(v[2], v[3])));
  }
#endif

  // wave32 + cross-wave max reduce
  #pragma unroll
  for (int off = 16; off > 0; off >>= 1)
    lmax = fmaxf(lmax, __shfl_xor(lmax, off, 32));
  __shared__ float wred[16];
  if ((tid & 31) == 0) wred[wave] = lmax;
  __syncthreads();
  if (tid == 0) {
    float r = wred[0];
    #pragma unroll
    for (int w = 1; w < 16; ++w) r = fmaxf(r, wred[w]);
    wmax_out[b] = r;
  }
}

// ---------------------------------------------------------------------------
// Kernel 3: scalar combine. out = mean(row_loss) + 0.1 * mean(wmax)
// ---------------------------------------------------------------------------
__global__ void finalize_kernel(const float* __restrict__ ws,
                                float* __restrict__ out) {
  const int tid = threadIdx.x;                         // 256 threads
  float l = ws[tid];
  float w = ws[BB + tid];
  #pragma unroll
  for (int off = 16; off > 0; off >>= 1) {
    l += __shfl_xor(l, off, 32);
    w += __shfl_xor(w, off, 32);
  }
  __shared__ float sl[8], sw[8];
  const int wave = tid >> 5;
  if ((tid & 31) == 0) { sl[wave] = l; sw[wave] = w; }
  __syncthreads();
  if (tid == 0) {
    float L = 0.0f, W = 0.0f;
    #pragma unroll
    for (int i = 0; i < 8; ++i) { L += sl[i]; W += sw[i]; }
    out[0] = L * (1.0f / (float)BB) + 0.1f * (W * (1.0f / (float)BB));
  }
}

// ---------------------------------------------------------------------------
extern "C" void kernel_launch(void* const* d_in, const int* in_sizes, int n_in,
                              void* d_out, int out_size, void* d_ws, size_t ws_size,
                              hipStream_t stream) {
  const float* logits = (const float*)d_in[0];   // [256, 32000] f32
  const float* weight = (const float*)d_in[1];   // [256, 512, 512] f32
  const int*   target = (const int*)d_in[2];     // [256] int
  float* ws  = (float*)d_ws;                     // ws[0..255]=row loss, ws[256..511]=wmax
  float* out = (float*)d_out;

  ce_kernel<<<BB, 256, 0, stream>>>(logits, target, ws);
  wmax_kernel<<<BB, 512, 2 * WTILE_ELEMS * sizeof(float), stream>>>(weight, ws + BB);
  finalize_kernel<<<1, 256, 0, stream>>>(ws, out);
}